// VectorQuantizer_66769561583982
// MI455X (gfx1250) — compile-verified
//
#include <hip/hip_runtime.h>

typedef __attribute__((ext_vector_type(2))) float v2f;
typedef __attribute__((ext_vector_type(8))) float v8f;

#define N_TOT 16384
#define NCB   8
#define DIM   32
#define CBS   1024

// w[k][c] = sum_d cb[k][c][d]^2 + rate_bias[k][c]
__global__ __launch_bounds__(256) void vq_prep(const float* __restrict__ cb,
                                               const float* __restrict__ bias,
                                               float* __restrict__ w) {
    int i = blockIdx.x * 256 + threadIdx.x;      // 0 .. 8191
    const float* p = cb + (size_t)i * DIM;
    float s = 0.f;
#pragma unroll
    for (int d = 0; d < DIM; ++d) s += p[d] * p[d];
    w[i] = s + bias[i];
}

// One workgroup = 16 rows of x, one codebook k. 256 threads = 8 waves.
__global__ __launch_bounds__(256) void vq_main(const float* __restrict__ x,
                                               const float* __restrict__ cb,
                                               const float* __restrict__ w,
                                               float* __restrict__ out) {
    __shared__ float xsq[16];
    __shared__ float candv[16][128];
    __shared__ int   candc[16][128];
    __shared__ int   bestc_s[16];

    const int nt   = blockIdx.x >> 3;
    const int k    = blockIdx.x & 7;
    const int n0   = nt << 4;
    const int t    = threadIdx.x;
    const int lane = t & 31;
    const int wave = t >> 5;
    const int lrow = t & 15;         // lane % 16
    const int hi   = lane >> 4;      // 0 or 1 (lane half)

    // ---- x row norms into LDS ----
    if (t < 16) {
        const float* xr = x + ((size_t)(n0 + t) * NCB + k) * DIM;
        float s = 0.f;
#pragma unroll
        for (int d = 0; d < DIM; ++d) s += xr[d] * xr[d];
        xsq[t] = s;
    }
    __syncthreads();

    // ---- A fragments: 16x4 f32 per WMMA; lanes 0-15 hold K{0,1}, 16-31 hold K{2,3}
    const float* xr = x + ((size_t)(n0 + lrow) * NCB + k) * DIM + hi * 2;
    v2f afrag[8];
#pragma unroll
    for (int kk = 0; kk < 8; ++kk) {
        afrag[kk].x = xr[kk * 4 + 0];
        afrag[kk].y = xr[kk * 4 + 1];
    }

    float myxsq[8];
#pragma unroll
    for (int i = 0; i < 8; ++i) myxsq[i] = xsq[i + hi * 8];

    float minv[8]; int minc[8];
#pragma unroll
    for (int i = 0; i < 8; ++i) { minv[i] = 3.4e38f; minc[i] = 0; }

    const float* cbk    = cb + (size_t)k * CBS * DIM;
    const float* cbbase = cbk + hi * 2;
    const float* wk     = w + k * CBS;

    // ---- software-pipelined sweep of 8 column tiles (16 codewords each) ----
    v2f bfA[8], bfB[8];
    float wcA, wcB;

    auto loadTile = [&](int j, v2f bf[8], float& wc) {
        const int c = ((wave * 8 + j) << 4) + lrow;
        const float* cr = cbbase + (size_t)c * DIM;
#pragma unroll
        for (int kk = 0; kk < 8; ++kk) {
            bf[kk].x = cr[kk * 4 + 0];
            bf[kk].y = cr[kk * 4 + 1];
        }
        wc = wk[c];
    };

    loadTile(0, bfA, wcA);

#pragma unroll
    for (int j = 0; j < 8; ++j) {
        v2f*  bf = (j & 1) ? bfB : bfA;
        float wc = (j & 1) ? wcB : wcA;     // copy out before prefetch overwrites

        if (j < 7) {                        // prefetch next tile while WMMAs run
            if (j & 1) loadTile(j + 1, bfA, wcA);
            else       loadTile(j + 1, bfB, wcB);
        }

        v8f acc = {0.f, 0.f, 0.f, 0.f, 0.f, 0.f, 0.f, 0.f};
#pragma unroll
        for (int kk = 0; kk < 8; ++kk) {
            acc = __builtin_amdgcn_wmma_f32_16x16x4_f32(
                /*neg_a=*/false, afrag[kk], /*neg_b=*/false, bf[kk],
                /*c_mod=*/(short)0, acc, /*reuse_a=*/false, /*reuse_b=*/false);
        }

        const int c = ((wave * 8 + j) << 4) + lrow;
#pragma unroll
        for (int i = 0; i < 8; ++i) {
            float d = fmaf(-2.f, acc[i], myxsq[i] + wc);
            if (d < minv[i]) { minv[i] = d; minc[i] = c; }
        }
    }

    // ---- stage per-lane argmin candidates ----
#pragma unroll
    for (int i = 0; i < 8; ++i) {
        const int row = i + hi * 8;
        candv[row][wave * 16 + lrow] = minv[i];
        candc[row][wave * 16 + lrow] = minc[i];
    }
    __syncthreads();

    const size_t ohOff  = (size_t)N_TOT * NCB * DIM;
    const size_t idxOff = ohOff + (size_t)N_TOT * NCB * CBS;

    // ---- final reduction: one thread per row over 128 candidates ----
    if (t < 16) {
        float bv = candv[t][0]; int bc = candc[t][0];
        for (int s = 1; s < 128; ++s) {
            float v = candv[t][s]; int c = candc[t][s];
            if (v < bv || (v == bv && c < bc)) { bv = v; bc = c; }
        }
        bestc_s[t] = bc;
        out[idxOff + (size_t)(n0 + t) * NCB + k] = (float)bc;
    }
    __syncthreads();

    // ---- one_hot block: 16 rows x 1024 cols, float4 stores (bandwidth pass) ----
    float* oh = out + ohOff;
#pragma unroll
    for (int i = 0; i < 16; ++i) {
        const int q   = t + (i << 8);        // float4 slot 0..4095
        const int row = q >> 8;              // 256 float4 per row
        const int c4  = (q & 255) << 2;
        const int bc  = bestc_s[row];
        float4 v;
        v.x = (c4 + 0 == bc) ? 1.f : 0.f;
        v.y = (c4 + 1 == bc) ? 1.f : 0.f;
        v.z = (c4 + 2 == bc) ? 1.f : 0.f;
        v.w = (c4 + 3 == bc) ? 1.f : 0.f;
        float* dst = oh + ((size_t)(n0 + row) * NCB + k) * CBS + c4;
        *(float4*)dst = v;
    }

    // ---- x_hat: gather selected codeword, 16 rows x 32 dims ----
#pragma unroll
    for (int i = 0; i < 2; ++i) {
        const int e   = t + (i << 8);
        const int row = e >> 5;
        const int d   = e & 31;
        out[((size_t)(n0 + row) * NCB + k) * DIM + d] =
            cbk[(size_t)bestc_s[row] * DIM + d];
    }
}

extern "C" void kernel_launch(void* const* d_in, const int* in_sizes, int n_in,
                              void* d_out, int out_size, void* d_ws, size_t ws_size,
                              hipStream_t stream) {
    (void)in_sizes; (void)n_in; (void)out_size; (void)ws_size;
    const float* x    = (const float*)d_in[0];
    const float* cb   = (const float*)d_in[1];
    const float* bias = (const float*)d_in[2];
    float* w = (float*)d_ws;   // 8192 floats = 32 KB scratch

    vq_prep<<<(NCB * CBS) / 256, 256, 0, stream>>>(cb, bias, w);
    vq_main<<<(N_TOT / 16) * NCB, 256, 0, stream>>>(x, cb, w, (float*)d_out);
}